// SpatialGraphConvBlock_49864570306621
// MI455X (gfx1250) — compile-verified
//
#include <hip/hip_runtime.h>
#include <hip/hip_bf16.h>

#define N_NODES 40000
#define N_EDGES 640000
#define D 128
#define LN_EPS 1e-5f

typedef float v2f __attribute__((ext_vector_type(2)));
typedef float v8f __attribute__((ext_vector_type(8)));

// ---------------- zero workspace ----------------
__global__ void zero_kernel(float* __restrict__ p, long long n) {
    long long i = (long long)blockIdx.x * blockDim.x + threadIdx.x;
    if (i < n) p[i] = 0.0f;
}

// ---------------- degree accumulation ----------------
__global__ void deg_kernel(const long long* __restrict__ row,
                           const float* __restrict__ w,
                           float* __restrict__ deg, int nE) {
    int e = blockIdx.x * blockDim.x + threadIdx.x;
    if (e >= nE) return;
    atomicAdd(deg + row[e], w[e]);
}

// ---------------- symmetric normalization ----------------
__global__ void norm_kernel(const long long* __restrict__ row,
                            const long long* __restrict__ col,
                            const float* __restrict__ w,
                            const float* __restrict__ deg,
                            float* __restrict__ nrm, int nE) {
    int e = blockIdx.x * blockDim.x + threadIdx.x;
    if (e >= nE) return;
    float dr = deg[row[e]];
    float dc = deg[col[e]];
    float ir = dr > 0.0f ? rsqrtf(dr) : 0.0f;
    float ic = dc > 0.0f ? rsqrtf(dc) : 0.0f;
    // scale = 2/lambda_max = 1 ; L_hat off-diag = -dinv[r]*w*dinv[c], diag term = 0
    nrm[e] = -ir * w[e] * ic;
}

// ---------------- sparse propagation: acc[row] += nrm * h[col] ----------------
// one wave32 per edge; each lane carries 4 channels (32*4 = 128)
__global__ __launch_bounds__(256) void prop_kernel(const float* __restrict__ h,
                                                   const long long* __restrict__ row,
                                                   const long long* __restrict__ col,
                                                   const float* __restrict__ nrm,
                                                   float* __restrict__ acc, int nE) {
    int e = blockIdx.x * 8 + (threadIdx.x >> 5);
    if (e >= nE) return;
    int lane = threadIdx.x & 31;
    long long r = row[e];
    long long c = col[e];
    float s = nrm[e];
    const float4* src = (const float4*)(h + c * (long long)D);
    float4 v = src[lane];
    float* dst = acc + r * (long long)D + lane * 4;
    atomicAdd(dst + 0, s * v.x);
    atomicAdd(dst + 1, s * v.y);
    atomicAdd(dst + 2, s * v.z);
    atomicAdd(dst + 3, s * v.w);
}

// ---------------- T2 = 2*prop(T1) - T0 ----------------
__global__ void combine_kernel(float* __restrict__ t2, const float* __restrict__ x, long long n) {
    long long i = (long long)blockIdx.x * blockDim.x + threadIdx.x;
    if (i < n) t2[i] = 2.0f * t2[i] - x[i];
}

// ---------------- WMMA tile helper: C += A[16xK] * B[Kx16] over K=128 ----------------
__device__ __forceinline__ v8f gemm_tile(const float* __restrict__ A,
                                         const float* __restrict__ B,
                                         int am, int bn, int lhalf, v8f c) {
    // 32-bit A 16x4 layout: lanes 0-15: VGPR0=K0,VGPR1=K1 ; lanes 16-31: VGPR0=K2,VGPR1=K3
    // B 4x16 mirrors the same K-per-lane-half assignment; rows striped across lanes.
    #pragma unroll
    for (int kk = 0; kk < D; kk += 4) {
        int ak = kk + lhalf * 2;
        v2f a;
        a.x = A[(long long)am * D + ak];
        a.y = A[(long long)am * D + ak + 1];
        v2f b;
        b.x = B[(long long)ak * D + bn];
        b.y = B[(long long)(ak + 1) * D + bn];
        c = __builtin_amdgcn_wmma_f32_16x16x4_f32(false, a, false, b, (short)0, c, false, false);
    }
    return c;
}

// ---------------- fused 3-hop GEMM + bias + LayerNorm + ReLU ----------------
// block = 16-row band; 8 waves cover the 8 column tiles of 16
__global__ __launch_bounds__(256) void gemm_ln_kernel(const float* __restrict__ T0,
                                                      const float* __restrict__ T1,
                                                      const float* __restrict__ T2,
                                                      const float* __restrict__ W,
                                                      const float* __restrict__ bias,
                                                      const float* __restrict__ gamma,
                                                      const float* __restrict__ beta,
                                                      float* __restrict__ out) {
    __shared__ float sh[16 * D];
    int tid = threadIdx.x;
    int wave = tid >> 5;           // column tile 0..7
    int lane = tid & 31;
    int lhalf = lane >> 4;         // 0 | 1
    int l16 = lane & 15;
    int m0 = blockIdx.x * 16;      // N = 2500 * 16 exactly -> no bounds checks, EXEC all ones
    int n0 = wave * 16;

    int am = m0 + l16;
    int bn = n0 + l16;

    v8f c = {0.f, 0.f, 0.f, 0.f, 0.f, 0.f, 0.f, 0.f};
    c = gemm_tile(T0, W,             am, bn, lhalf, c);
    c = gemm_tile(T1, W + D * D,     am, bn, lhalf, c);
    c = gemm_tile(T2, W + 2 * D * D, am, bn, lhalf, c);

    // C/D layout: VGPR r -> row r (lanes 0-15) / row r+8 (lanes 16-31), col = lane%16
    #pragma unroll
    for (int r = 0; r < 8; ++r) {
        int m = r + lhalf * 8;
        int n = n0 + l16;
        sh[m * D + n] = c[r] + bias[n];
    }
    __syncthreads();

    // LayerNorm + ReLU: 16 rows, each handled by a 16-lane group, 8 channels/lane
    int rowLocal = tid >> 4;       // 0..15
    int g = tid & 15;
    const float* rp = sh + rowLocal * D;
    float s = 0.f, sq = 0.f;
    #pragma unroll
    for (int i = 0; i < 8; ++i) {
        float v = rp[g * 8 + i];
        s += v;
        sq += v * v;
    }
    #pragma unroll
    for (int off = 8; off >= 1; off >>= 1) {
        s  += __shfl_xor(s, off, 16);
        sq += __shfl_xor(sq, off, 16);
    }
    float mean = s * (1.0f / D);
    float var  = sq * (1.0f / D) - mean * mean;
    float inv  = rsqrtf(var + LN_EPS);
    float* op = out + (long long)(m0 + rowLocal) * D;
    #pragma unroll
    for (int i = 0; i < 8; ++i) {
        int cidx = g * 8 + i;
        float v = (rp[cidx] - mean) * inv * gamma[cidx] + beta[cidx];
        op[cidx] = fmaxf(v, 0.0f);
    }
}

extern "C" void kernel_launch(void* const* d_in, const int* in_sizes, int n_in,
                              void* d_out, int out_size, void* d_ws, size_t ws_size,
                              hipStream_t stream) {
    const float*     x    = (const float*)d_in[0];        // [N,128]
    const float*     ew   = (const float*)d_in[1];        // [E]
    const float*     W    = (const float*)d_in[2];        // [3,128,128]
    const float*     bias = (const float*)d_in[3];        // [128]
    const float*     gam  = (const float*)d_in[4];        // [128]
    const float*     bet  = (const float*)d_in[5];        // [128]
    const long long* ei   = (const long long*)d_in[6];    // [2,E] int64
    const long long* row  = ei;
    const long long* col  = ei + N_EDGES;
    float* out = (float*)d_out;

    // workspace layout: [deg N][T1 N*128][T2 N*128][norm E]
    float* deg = (float*)d_ws;
    float* T1  = deg + N_NODES;
    float* T2  = T1 + (long long)N_NODES * D;
    float* nrm = T2 + (long long)N_NODES * D;

    // zero the atomic accumulators (deg,T1,T2 are contiguous)
    long long nz = (long long)N_NODES + 2LL * N_NODES * D;
    zero_kernel<<<(unsigned)((nz + 255) / 256), 256, 0, stream>>>(deg, nz);

    deg_kernel<<<(N_EDGES + 255) / 256, 256, 0, stream>>>(row, ew, deg, N_EDGES);
    norm_kernel<<<(N_EDGES + 255) / 256, 256, 0, stream>>>(row, col, ew, deg, nrm, N_EDGES);

    // T1 = L_hat x
    prop_kernel<<<(N_EDGES + 7) / 8, 256, 0, stream>>>(x, row, col, nrm, T1, N_EDGES);
    // T2 = 2 * L_hat T1 - x
    prop_kernel<<<(N_EDGES + 7) / 8, 256, 0, stream>>>(T1, row, col, nrm, T2, N_EDGES);
    long long nel = (long long)N_NODES * D;
    combine_kernel<<<(unsigned)((nel + 255) / 256), 256, 0, stream>>>(T2, x, nel);

    // out = relu(LN(T0@W0 + T1@W1 + T2@W2 + bias))
    gemm_ln_kernel<<<N_NODES / 16, 256, 0, stream>>>(x, T1, T2, W, bias, gam, bet, out);
}